// WaveNetStep_69252052680869
// MI455X (gfx1250) — compile-verified
//
#include <hip/hip_runtime.h>
#include <math.h>

typedef float v2f __attribute__((ext_vector_type(2)));
typedef float v8f __attribute__((ext_vector_type(8)));

#define B_TOT 2048
#define CH    64
#define SKIP  96
#define TLEN  1023
#define NBLK  10
#define RES_SCALE 0.3f

__device__ inline v8f wmma4(v2f a, v2f b, v8f c) {
  // D = A(16x4) * B(4x16) + C(16x16), all f32. 8 args: neg_a,A,neg_b,B,c_mod,C,reuse_a,reuse_b
  return __builtin_amdgcn_wmma_f32_16x16x4_f32(false, a, false, b, (short)0, c, false, false);
}

__device__ inline v8f splat8(float v) {
  v8f r;
#pragma unroll
  for (int i = 0; i < 8; ++i) r[i] = v;
  return r;
}

// ---------------------------------------------------------------------------
// Kernel 1: recurrent chain + head. One wave (= one workgroup) = 16 batch rows.
// grid = 128 blocks x 32 threads -> 128 independent waves spread across WGPs.
// Writes: out[B,2] and cur_cols[i][b][c] (snapshot of `current` entering block i).
// ---------------------------------------------------------------------------
__global__ __launch_bounds__(32)
void wavenet_chain(const float* __restrict__ x,      // (B,32)
                   const float* __restrict__ buf,    // (B,64,1023)
                   const float* __restrict__ inp_w,  // (64,32)
                   const float* __restrict__ inp_b,  // (64)
                   const float* __restrict__ fw0,    // (10,64,64)
                   const float* __restrict__ fw1,
                   const float* __restrict__ fb,     // (10,64)
                   const float* __restrict__ gw0,
                   const float* __restrict__ gw1,
                   const float* __restrict__ gb,
                   const float* __restrict__ rw,     // (10,64,64)
                   const float* __restrict__ rb,     // (10,64)
                   const float* __restrict__ sw,     // (10,96,64)
                   const float* __restrict__ sb,     // (10,96)
                   const float* __restrict__ h1w,    // (96,96)
                   const float* __restrict__ h1b,    // (96)
                   const float* __restrict__ h2w,    // (2,96)
                   const float* __restrict__ h2b,    // (2)
                   float* __restrict__ out,          // (B,2)
                   float* __restrict__ cur_cols)     // (10,B,64) staging in d_ws
{
  __shared__ float lds[3584];       // cur(16x64) | z(16x64) | sk(16x96)
  const int lane = threadIdx.x & 31;
  const int half = lane >> 4;       // 0..1
  const int idx  = lane & 15;       // 0..15
  const int b0   = blockIdx.x * 16;

  float* curL = &lds[0];
  float* zL   = curL + 1024;
  float* skL  = curL + 2048;

  // ---- input layer: current = x @ inp_w.T + inp_b  (M=16, K=32, N=64)
  {
    v8f acc[4];
#pragma unroll
    for (int nt = 0; nt < 4; ++nt) acc[nt] = splat8(inp_b[nt * 16 + idx]);
#pragma unroll
    for (int ks = 0; ks < 8; ++ks) {
      const int kk = ks * 4 + half * 2;
      v2f a = *(const v2f*)(x + (size_t)(b0 + idx) * 32 + kk);
#pragma unroll
      for (int nt = 0; nt < 4; ++nt) {
        v2f bm = *(const v2f*)(inp_w + (size_t)(nt * 16 + idx) * 32 + kk);
        acc[nt] = wmma4(a, bm, acc[nt]);
      }
    }
#pragma unroll
    for (int nt = 0; nt < 4; ++nt)
#pragma unroll
      for (int v = 0; v < 8; ++v)
        curL[(v + half * 8) * 64 + nt * 16 + idx] = acc[nt][v];
  }
  __syncthreads();

  // persistent skip accumulators (16 x 96)
  v8f skA[6];
#pragma unroll
  for (int nt = 0; nt < 6; ++nt) skA[nt] = splat8(0.0f);

#pragma unroll 1
  for (int i = 0; i < NBLK; ++i) {
    const int off = (1 << i) - 1;   // OFFSETS[i]

    // snapshot `current` (pre-residual) -> cur_cols[i][b][c], fully coalesced
#pragma unroll
    for (int j = 0; j < 32; ++j) {
      const int e = j * 32 + lane;            // 0..1023
      const int m = e >> 6, c = e & 63;
      cur_cols[(size_t)i * (B_TOT * CH) + (size_t)(b0 + m) * CH + c] = curL[m * 64 + c];
    }

    // ---- fused f|g GEMM: [delayed|current](16x128) @ [W*0|W*1].T, N=128
    v8f fg[8];
#pragma unroll
    for (int nt = 0; nt < 8; ++nt)
      fg[nt] = splat8((nt < 4) ? fb[i * 64 + nt * 16 + idx]
                               : gb[i * 64 + (nt - 4) * 16 + idx]);

    const float* dBase = buf + (size_t)(b0 + idx) * (CH * TLEN) + off;
    const float* w0f = fw0 + (size_t)i * 4096;
    const float* w0g = gw0 + (size_t)i * 4096;
    const float* w1f = fw1 + (size_t)i * 4096;
    const float* w1g = gw1 + (size_t)i * 4096;

#pragma unroll
    for (int ks = 0; ks < 16; ++ks) {         // K part 1: A = delayed (gather)
      const int kk = ks * 4 + half * 2;
      v2f a;
      a.x = dBase[(size_t)kk * TLEN];
      a.y = dBase[(size_t)(kk + 1) * TLEN];
#pragma unroll
      for (int nt = 0; nt < 8; ++nt) {
        const float* wsel = (nt < 4) ? w0f + (size_t)(nt * 16 + idx) * 64
                                     : w0g + (size_t)((nt - 4) * 16 + idx) * 64;
        v2f bm = *(const v2f*)(wsel + kk);
        fg[nt] = wmma4(a, bm, fg[nt]);
      }
    }
#pragma unroll
    for (int ks = 0; ks < 16; ++ks) {         // K part 2: A = current (LDS)
      const int kk = ks * 4 + half * 2;
      v2f a = *(const v2f*)(curL + idx * 64 + kk);
#pragma unroll
      for (int nt = 0; nt < 8; ++nt) {
        const float* wsel = (nt < 4) ? w1f + (size_t)(nt * 16 + idx) * 64
                                     : w1g + (size_t)((nt - 4) * 16 + idx) * 64;
        v2f bm = *(const v2f*)(wsel + kk);
        fg[nt] = wmma4(a, bm, fg[nt]);
      }
    }

    // ---- z = tanh(f) * sigmoid(g)  -> zL (row-major 16x64)
#pragma unroll
    for (int nt = 0; nt < 4; ++nt)
#pragma unroll
      for (int v = 0; v < 8; ++v) {
        const float f = fg[nt][v], g = fg[nt + 4][v];
        const float z = tanhf(f) * (1.0f / (1.0f + __expf(-g)));
        zL[(v + half * 8) * 64 + nt * 16 + idx] = z;
      }
    __syncthreads();

    // ---- fused skip (N=96, into skA) + res (N=64) GEMM over A = z
    v8f rA[4];
#pragma unroll
    for (int nt = 0; nt < 4; ++nt) rA[nt] = splat8(rb[i * 64 + nt * 16 + idx]);
#pragma unroll
    for (int nt = 0; nt < 6; ++nt) {          // add skip_b[i] once per block
      const float sbv = sb[i * 96 + nt * 16 + idx];
#pragma unroll
      for (int v = 0; v < 8; ++v) skA[nt][v] += sbv;
    }
#pragma unroll
    for (int ks = 0; ks < 16; ++ks) {
      const int kk = ks * 4 + half * 2;
      v2f a = *(const v2f*)(zL + idx * 64 + kk);
#pragma unroll
      for (int nt = 0; nt < 6; ++nt) {
        v2f bm = *(const v2f*)(sw + (size_t)i * (SKIP * CH) + (size_t)(nt * 16 + idx) * 64 + kk);
        skA[nt] = wmma4(a, bm, skA[nt]);
      }
#pragma unroll
      for (int nt = 0; nt < 4; ++nt) {
        v2f bm = *(const v2f*)(rw + (size_t)i * 4096 + (size_t)(nt * 16 + idx) * 64 + kk);
        rA[nt] = wmma4(a, bm, rA[nt]);
      }
    }
    __syncthreads();

    // ---- current += RES_SCALE * res
#pragma unroll
    for (int nt = 0; nt < 4; ++nt)
#pragma unroll
      for (int v = 0; v < 8; ++v)
        curL[(v + half * 8) * 64 + nt * 16 + idx] += RES_SCALE * rA[nt][v];
    __syncthreads();
  }

  // ---- head: h0 = relu(skip_sum) -> skL (16x96)
#pragma unroll
  for (int nt = 0; nt < 6; ++nt)
#pragma unroll
    for (int v = 0; v < 8; ++v)
      skL[(v + half * 8) * 96 + nt * 16 + idx] = fmaxf(skA[nt][v], 0.0f);
  __syncthreads();

  // h1 = relu(h0 @ head1_w.T + head1_b): M=16, K=96, N=96
  v8f hA[6];
#pragma unroll
  for (int nt = 0; nt < 6; ++nt) hA[nt] = splat8(h1b[nt * 16 + idx]);
#pragma unroll
  for (int ks = 0; ks < 24; ++ks) {
    const int kk = ks * 4 + half * 2;
    v2f a = *(const v2f*)(skL + idx * 96 + kk);
#pragma unroll
    for (int nt = 0; nt < 6; ++nt) {
      v2f bm = *(const v2f*)(h1w + (size_t)(nt * 16 + idx) * 96 + kk);
      hA[nt] = wmma4(a, bm, hA[nt]);
    }
  }
  __syncthreads();
#pragma unroll
  for (int nt = 0; nt < 6; ++nt)
#pragma unroll
    for (int v = 0; v < 8; ++v)
      skL[(v + half * 8) * 96 + nt * 16 + idx] = fmaxf(hA[nt][v], 0.0f);
  __syncthreads();

  // out = h1 @ head2_w.T + head2_b : N=2, per-lane dot (m=idx, n=half)
  {
    float s = h2b[half];
#pragma unroll
    for (int j = 0; j < 96; ++j) s += skL[idx * 96 + j] * h2w[half * 96 + j];
    out[(size_t)(b0 + idx) * 2 + half] = s;
  }
}

// ---------------------------------------------------------------------------
// Kernel 2: streaming shift-copy + snapshot merge (HBM-bound, ~1.07 GB).
// new[b,c,k] = old[b,c,k+1], except k with (k+2) a power of two, which takes
// the block-i snapshot (i = log2(k+2)-1) from cur_cols.
// ---------------------------------------------------------------------------
__global__ __launch_bounds__(256)
void wavenet_shift(const float* __restrict__ buf,
                   const float* __restrict__ cur_cols,   // (10,B,64)
                   float* __restrict__ nbuf)
{
  const size_t total = (size_t)B_TOT * CH * TLEN;
  size_t idx = (size_t)blockIdx.x * 256 + threadIdx.x;
  if (idx >= total) return;
  const int k = (int)(idx % TLEN);
  if (((k + 2) & (k + 1)) == 0) {
    const int i = 30 - __clz(k + 2);                  // log2(k+2) - 1
    const size_t row = idx / TLEN;                    // b*64 + c
    nbuf[idx] = cur_cols[(size_t)i * (B_TOT * CH) + row];
  } else {
    nbuf[idx] = buf[idx + 1];
  }
}

extern "C" void kernel_launch(void* const* d_in, const int* in_sizes, int n_in,
                              void* d_out, int out_size, void* d_ws, size_t ws_size,
                              hipStream_t stream) {
  const float* x     = (const float*)d_in[0];
  const float* buf   = (const float*)d_in[1];
  const float* inp_w = (const float*)d_in[2];
  const float* inp_b = (const float*)d_in[3];
  const float* fw0   = (const float*)d_in[4];
  const float* fw1   = (const float*)d_in[5];
  const float* fb    = (const float*)d_in[6];
  const float* gw0   = (const float*)d_in[7];
  const float* gw1   = (const float*)d_in[8];
  const float* gb    = (const float*)d_in[9];
  const float* rw    = (const float*)d_in[10];
  const float* rb    = (const float*)d_in[11];
  const float* sw    = (const float*)d_in[12];
  const float* sb    = (const float*)d_in[13];
  const float* h1w   = (const float*)d_in[14];
  const float* h1b   = (const float*)d_in[15];
  const float* h2w   = (const float*)d_in[16];
  const float* h2b   = (const float*)d_in[17];

  float* out  = (float*)d_out;                        // (B,2)
  float* nbuf = (float*)d_out + (size_t)B_TOT * 2;    // (B,64,1023)
  float* cur_cols = (float*)d_ws;                     // (10,B,64) = 5.24 MB

  wavenet_chain<<<B_TOT / 16, 32, 0, stream>>>(
      x, buf, inp_w, inp_b, fw0, fw1, fb, gw0, gw1, gb,
      rw, rb, sw, sb, h1w, h1b, h2w, h2b, out, cur_cols);

  const size_t total = (size_t)B_TOT * CH * TLEN;
  const int blocks = (int)((total + 255) / 256);
  wavenet_shift<<<blocks, 256, 0, stream>>>(buf, cur_cols, nbuf);
}